// HeteroGraphormerLayerComplete_16252156248337
// MI455X (gfx1250) — compile-verified
//
#include <hip/hip_runtime.h>
#include <hip/hip_bf16.h>
#include <math.h>

#define Nn 4096
#define Cc 256
#define Hh 8
#define DHd 32
#define Ee 32768
#define MAXSPD 3
#define FWORDS 128   // 4096 bits / 32

typedef float v2f __attribute__((ext_vector_type(2)));
typedef float v8f __attribute__((ext_vector_type(8)));

// ---------------- small helpers ----------------

__device__ __forceinline__ void atomicMaxF(float* addr, float val) {
    unsigned int* ua = (unsigned int*)addr;
    unsigned int old = __hip_atomic_load(ua, __ATOMIC_RELAXED, __HIP_MEMORY_SCOPE_AGENT);
    while (true) {
        float cur = __uint_as_float(old);
        if (cur >= val) break;
        unsigned int assumed = old;
        old = atomicCAS(ua, assumed, __float_as_uint(val));
        if (old == assumed) break;
    }
}

__device__ __forceinline__ int deg_bucket(float d) {
    int b = (int)floorf(log2f(d + 1.0f));
    if (b < 0) b = 0;
    if (b > 8) b = 8;
    return b;
}

// ---------------- degree + input embedding ----------------

__global__ void deg_kernel(const int* __restrict__ src, const int* __restrict__ dst,
                           unsigned* __restrict__ indeg, unsigned* __restrict__ outdeg) {
    int e = blockIdx.x * blockDim.x + threadIdx.x;
    if (e >= Ee) return;
    atomicAdd(&outdeg[src[e]], 1u);
    atomicAdd(&indeg[dst[e]], 1u);
}

__global__ void xin_kernel(const float* __restrict__ x,
                           const unsigned* __restrict__ indeg, const unsigned* __restrict__ outdeg,
                           const float* __restrict__ inEmb, const float* __restrict__ outEmb,
                           float* __restrict__ x_in) {
    int i = blockIdx.x;          // node
    int c = threadIdx.x;         // channel
    int bi = deg_bucket((float)indeg[i]);
    int bo = deg_bucket((float)outdeg[i]);
    x_in[i * Cc + c] = x[i * Cc + c] + inEmb[bi * Cc + c] + outEmb[bo * Cc + c];
}

// ---------------- SPD BFS over bitset rows ----------------

__global__ void f_ident_kernel(unsigned* __restrict__ F) {
    int i = blockIdx.x * blockDim.x + threadIdx.x;
    if (i >= Nn) return;
    F[i * FWORDS + (i >> 5)] = (1u << (i & 31));
}

__global__ void bfs_scatter(const unsigned* __restrict__ Fold, unsigned* __restrict__ Fnew,
                            const int* __restrict__ src, const int* __restrict__ dst) {
    int e = blockIdx.x;          // one block per edge
    int w = threadIdx.x;         // 128 words
    int s = src[e], d = dst[e];
    unsigned v = Fold[s * FWORDS + w];
    if (v) atomicOr(&Fnew[d * FWORDS + w], v);
}

__global__ void bfs_probe(const unsigned* __restrict__ F,
                          const int* __restrict__ src, const int* __restrict__ dst,
                          int* __restrict__ spd, int k) {
    int e = blockIdx.x * blockDim.x + threadIdx.x;
    if (e >= Ee) return;
    if (spd[e] != 0) return;
    int d = dst[e];
    unsigned v = F[src[e] * FWORDS + (d >> 5)];
    if (v & (1u << (d & 31))) spd[e] = k;
}

// ---------------- LayerNorm (one block = one node, 256 threads) ----------------

__global__ void ln_kernel(const float* __restrict__ in, const float* __restrict__ g,
                          const float* __restrict__ b, float* __restrict__ out) {
    __shared__ float s1[Cc];
    __shared__ float s2[Cc];
    int row = blockIdx.x, c = threadIdx.x;
    float v = in[row * Cc + c];
    s1[c] = v;
    s2[c] = v * v;
    __syncthreads();
    for (int off = Cc / 2; off > 0; off >>= 1) {
        if (c < off) { s1[c] += s1[c + off]; s2[c] += s2[c + off]; }
        __syncthreads();
    }
    float mu  = s1[0] * (1.0f / Cc);
    float var = s2[0] * (1.0f / Cc) - mu * mu;
    float inv = rsqrtf(var + 1e-5f);
    out[row * Cc + c] = g[c] * (v - mu) * inv + b[c];
}

// ---------------- LDS-staged f32 WMMA GEMM ----------------
// C = act(A@B + bias) [+ residual], A:[M,K] B:[K,N] row-major.
// Workgroup: 256 threads = 8 waves; tile 32(M) x 64(N); K-chunk 32.
// Each wave owns one 16x16 tile: waves laid out 2(M) x 4(N).
// Fragment layouts per CDNA5 ISA §7.12.2 (V_WMMA_F32_16X16X4_F32):
//   A 16x4 : row = lane&15; lane half selects K pair -> g=(lane>>4)*2
//   B 4x16 : col = lane&15; same K half-grouping
//   C 16x16: VGPR r -> row r + 8*(lane>>4), col lane&15
// LDS pads: A stride 36 (16B-aligned float4 stores; rows 0..15 map to 16
// distinct banks since 36*16 = 9*64); B stride 72 (k vs k+2 rows land 16
// banks apart -> lane halves conflict-free).

#define BM 32
#define BN 64
#define BK 32
#define APAD 36
#define BPAD 72

__global__ __launch_bounds__(256)
void gemm_wmma_f32(const float* __restrict__ A, const float* __restrict__ B,
                   const float* __restrict__ bias, const float* __restrict__ residual,
                   float* __restrict__ Cmat, int M, int Ncols, int K, int gelu) {
    __shared__ float Alds[BM * APAD];
    __shared__ float Blds[BK * BPAD];

    const int tid  = threadIdx.x;
    const int lane = tid & 31;
    const int wid  = tid >> 5;           // 0..7
    const int tm2  = wid >> 2;           // 0..1  (M sub-tile)
    const int tn2  = wid & 3;            // 0..3  (N sub-tile)

    const int rowBase = blockIdx.y * BM;
    const int colBase = blockIdx.x * BN;

    // cooperative stage indices
    const int ar = tid >> 3;             // 0..31
    const int ac = (tid & 7) << 2;       // 0..28 step 4
    const int br = tid >> 3;             // 0..31
    const int bc = (tid & 7) << 3;       // 0..56 step 8

    const int g  = (lane >> 4) << 1;     // 0 | 2 (K half-group)
    const int ml = tm2 * 16 + (lane & 15);
    const int nl = tn2 * 16 + (lane & 15);

    v8f acc = {};

    for (int k0 = 0; k0 < K; k0 += BK) {
        // ---- stage A (32x32) and B (32x64) via b128 loads ----
        const float4 av  = *(const float4*)(A + (size_t)(rowBase + ar) * K + k0 + ac);
        const float4 bv0 = *(const float4*)(B + (size_t)(k0 + br) * Ncols + colBase + bc);
        const float4 bv1 = *(const float4*)(B + (size_t)(k0 + br) * Ncols + colBase + bc + 4);
        *(float4*)&Alds[ar * APAD + ac] = av;
        *(float4*)&Blds[br * BPAD + bc] = bv0;
        *(float4*)&Blds[br * BPAD + bc + 4] = bv1;

        // prefetch next chunk while this one is consumed (global_prefetch_b8)
        if (k0 + BK < K) {
            __builtin_prefetch(A + (size_t)(rowBase + ar) * K + k0 + BK + ac, 0, 1);
            __builtin_prefetch(B + (size_t)(k0 + BK + br) * Ncols + colBase + bc, 0, 1);
        }
        __syncthreads();

        // ---- 8 WMMA steps over the chunk, fragments from LDS ----
#pragma unroll
        for (int kk = 0; kk < BK; kk += 4) {
            v2f a, b;
            a.x = Alds[ml * APAD + kk + g];
            a.y = Alds[ml * APAD + kk + g + 1];
            b.x = Blds[(kk + g) * BPAD + nl];
            b.y = Blds[(kk + g + 1) * BPAD + nl];
            acc = __builtin_amdgcn_wmma_f32_16x16x4_f32(
                /*neg_a=*/false, a, /*neg_b=*/false, b,
                /*c_mod=*/(short)0, acc, /*reuse_a=*/false, /*reuse_b=*/false);
        }
        __syncthreads();
    }

    // ---- epilogue: bias (+ exact GELU) (+ residual) ----
    const int col   = colBase + nl;
    const int rbase = rowBase + tm2 * 16 + ((lane >> 4) << 3);
    const float bval = bias[col];
#pragma unroll
    for (int r = 0; r < 8; ++r) {
        int row = rbase + r;
        float v = acc[r] + bval;
        if (gelu) v = 0.5f * v * (1.0f + erff(v * 0.70710678118654752f));
        if (residual) v += residual[(size_t)row * Ncols + col];
        Cmat[(size_t)row * Ncols + col] = v;
    }
}

// ---------------- attention: scores, softmax, aggregation ----------------

__global__ void fillf_kernel(float* __restrict__ p, float v, int n) {
    int i = blockIdx.x * blockDim.x + threadIdx.x;
    if (i < n) p[i] = v;
}

__global__ void score_kernel(const float* __restrict__ Q, const float* __restrict__ K,
                             const int* __restrict__ src, const int* __restrict__ dst,
                             const int* __restrict__ spd, const float* __restrict__ spd_emb,
                             const float* __restrict__ et_bias,
                             float* __restrict__ scores, float* __restrict__ mmax) {
    int idx = blockIdx.x * blockDim.x + threadIdx.x;   // e*H + h
    if (idx >= Ee * Hh) return;
    int e = idx >> 3, h = idx & 7;
    int s = src[e], d = dst[e];
    const float4* q  = (const float4*)(Q + (size_t)d * Cc + h * DHd);
    const float4* kk = (const float4*)(K + (size_t)s * Cc + h * DHd);
    float acc = 0.0f;
#pragma unroll
    for (int i = 0; i < DHd / 4; ++i) {
        float4 qa = q[i], ka = kk[i];
        acc += qa.x * ka.x + qa.y * ka.y + qa.z * ka.z + qa.w * ka.w;
    }
    int sv = spd[e];
    if (sv == 0) sv = MAXSPD + 1;                       // unreached -> 4 (rows 1..4 of spd_emb)
    float sc = acc * 0.17677669529663687f               // 1/sqrt(32)
             + spd_emb[sv * Hh + h] + et_bias[0];
    scores[idx] = sc;
    atomicMaxF(&mmax[d * Hh + h], sc);
}

__global__ void expsum_kernel(const int* __restrict__ dst,
                              const float* __restrict__ mmax,
                              float* __restrict__ scores, float* __restrict__ ssum) {
    int idx = blockIdx.x * blockDim.x + threadIdx.x;
    if (idx >= Ee * Hh) return;
    int e = idx >> 3, h = idx & 7;
    int d = dst[e];
    float ev = expf(scores[idx] - mmax[d * Hh + h]);
    scores[idx] = ev;
    atomicAdd(&ssum[d * Hh + h], ev);
}

__global__ void message_kernel(const int* __restrict__ src, const int* __restrict__ dst,
                               const float* __restrict__ scores, const float* __restrict__ ssum,
                               const float* __restrict__ V, float* __restrict__ msg) {
    int e = blockIdx.x;          // one block per edge
    int c = threadIdx.x;         // channel = h*32 + d
    int h = c >> 5;
    int s = src[e], d = dst[e];
    float sum = ssum[d * Hh + h];
    sum = (sum > 0.0f) ? sum : 1.0f;
    float w = scores[e * Hh + h] / sum;
    atomicAdd(&msg[(size_t)d * Cc + c], w * V[(size_t)s * Cc + c]);
}

// ---------------- launcher ----------------

extern "C" void kernel_launch(void* const* d_in, const int* in_sizes, int n_in,
                              void* d_out, int out_size, void* d_ws, size_t ws_size,
                              hipStream_t stream) {
    const float* x       = (const float*)d_in[0];
    const int*   eidx    = (const int*)d_in[1];
    const float* Wq      = (const float*)d_in[2];
    const float* bq      = (const float*)d_in[3];
    const float* Wk      = (const float*)d_in[4];
    const float* bk      = (const float*)d_in[5];
    const float* Wv      = (const float*)d_in[6];
    const float* bv      = (const float*)d_in[7];
    const float* Wo      = (const float*)d_in[8];
    const float* bo      = (const float*)d_in[9];
    const float* W1      = (const float*)d_in[10];
    const float* b1      = (const float*)d_in[11];
    const float* W2      = (const float*)d_in[12];
    const float* b2      = (const float*)d_in[13];
    const float* g1      = (const float*)d_in[14];
    const float* beta1   = (const float*)d_in[15];
    const float* g2      = (const float*)d_in[16];
    const float* beta2   = (const float*)d_in[17];
    const float* spd_emb = (const float*)d_in[18];
    const float* inEmb   = (const float*)d_in[19];
    const float* outEmb  = (const float*)d_in[20];
    const float* et_bias = (const float*)d_in[21];

    const int* src = eidx;          // edge_index[0]
    const int* dst = eidx + Ee;     // edge_index[1]
    float* out = (float*)d_out;

    // ---- carve workspace ----
    char* p = (char*)d_ws;
    auto take = [&](size_t bytes) { char* r = p; p += (bytes + 255) & ~size_t(255); return r; };
    unsigned* indeg  = (unsigned*)take(Nn * 4);
    unsigned* outdeg = (unsigned*)take(Nn * 4);
    unsigned* F0     = (unsigned*)take((size_t)Nn * FWORDS * 4);   // 2 MB
    unsigned* F1     = (unsigned*)take((size_t)Nn * FWORDS * 4);   // 2 MB
    int*      spd    = (int*)take(Ee * 4);
    float*    x_in   = (float*)take((size_t)Nn * Cc * 4);
    float*    xn     = (float*)take((size_t)Nn * Cc * 4);
    float*    Qb     = (float*)take((size_t)Nn * Cc * 4);
    float*    Kb     = (float*)take((size_t)Nn * Cc * 4);
    float*    Vb     = (float*)take((size_t)Nn * Cc * 4);
    float*    scores = (float*)take((size_t)Ee * Hh * 4);
    float*    mmax   = (float*)take((size_t)Nn * Hh * 4);
    float*    ssum   = (float*)take((size_t)Nn * Hh * 4);
    float*    msg    = (float*)take((size_t)Nn * Cc * 4);
    float*    hbuf   = (float*)take((size_t)Nn * Cc * 4);
    float*    hn     = (float*)take((size_t)Nn * Cc * 4);
    float*    ff1    = (float*)take((size_t)Nn * 4 * Cc * 4);      // 16 MB

    // ---- zero-init (every call; graph-capture-safe async memsets) ----
    hipMemsetAsync(indeg, 0, Nn * 4, stream);
    hipMemsetAsync(outdeg, 0, Nn * 4, stream);
    hipMemsetAsync(F0, 0, (size_t)Nn * FWORDS * 4, stream);
    hipMemsetAsync(spd, 0, Ee * 4, stream);
    hipMemsetAsync(ssum, 0, (size_t)Nn * Hh * 4, stream);
    hipMemsetAsync(msg, 0, (size_t)Nn * Cc * 4, stream);

    // ---- degrees + input embedding ----
    deg_kernel<<<(Ee + 255) / 256, 256, 0, stream>>>(src, dst, indeg, outdeg);
    xin_kernel<<<Nn, Cc, 0, stream>>>(x, indeg, outdeg, inEmb, outEmb, x_in);

    // ---- SPD BFS: F_k[d] = OR_{s->d} F_{k-1}[s], probe F_k[src] bit dst ----
    f_ident_kernel<<<(Nn + 255) / 256, 256, 0, stream>>>(F0);
    unsigned* Fcur = F0;
    unsigned* Fnxt = F1;
    for (int k = 1; k <= MAXSPD; ++k) {
        hipMemsetAsync(Fnxt, 0, (size_t)Nn * FWORDS * 4, stream);
        bfs_scatter<<<Ee, FWORDS, 0, stream>>>(Fcur, Fnxt, src, dst);
        bfs_probe<<<(Ee + 255) / 256, 256, 0, stream>>>(Fnxt, src, dst, spd, k);
        unsigned* t = Fcur; Fcur = Fnxt; Fnxt = t;
    }

    // ---- LN1 + QKV projections (f32 WMMA) ----
    ln_kernel<<<Nn, Cc, 0, stream>>>(x_in, g1, beta1, xn);
    {
        dim3 grid(Cc / BN, Nn / BM);          // 4 x 128
        gemm_wmma_f32<<<grid, 256, 0, stream>>>(xn, Wq, bq, nullptr, Qb, Nn, Cc, Cc, 0);
        gemm_wmma_f32<<<grid, 256, 0, stream>>>(xn, Wk, bk, nullptr, Kb, Nn, Cc, Cc, 0);
        gemm_wmma_f32<<<grid, 256, 0, stream>>>(xn, Wv, bv, nullptr, Vb, Nn, Cc, Cc, 0);
    }

    // ---- attention: scores -> segment softmax -> aggregation ----
    fillf_kernel<<<(Nn * Hh + 255) / 256, 256, 0, stream>>>(mmax, -INFINITY, Nn * Hh);
    score_kernel<<<(Ee * Hh + 255) / 256, 256, 0, stream>>>(Qb, Kb, src, dst, spd, spd_emb,
                                                            et_bias, scores, mmax);
    expsum_kernel<<<(Ee * Hh + 255) / 256, 256, 0, stream>>>(dst, mmax, scores, ssum);
    message_kernel<<<Ee, Cc, 0, stream>>>(src, dst, scores, ssum, Vb, msg);

    // ---- h = x_in + msg @ Wo + bo ----
    {
        dim3 grid(Cc / BN, Nn / BM);
        gemm_wmma_f32<<<grid, 256, 0, stream>>>(msg, Wo, bo, x_in, hbuf, Nn, Cc, Cc, 0);
    }

    // ---- LN2 + FFN ----
    ln_kernel<<<Nn, Cc, 0, stream>>>(hbuf, g2, beta2, hn);
    {
        dim3 grid1((4 * Cc) / BN, Nn / BM);   // 16 x 128
        gemm_wmma_f32<<<grid1, 256, 0, stream>>>(hn, W1, b1, nullptr, ff1, Nn, 4 * Cc, Cc, 1);

        dim3 grid2(Cc / BN, Nn / BM);
        gemm_wmma_f32<<<grid2, 256, 0, stream>>>(ff1, W2, b2, hbuf, out, Nn, Cc, 4 * Cc, 0);
    }
}